// NAP_21861383537402
// MI455X (gfx1250) — compile-verified
//
#include <hip/hip_runtime.h>
#include <hip/hip_bf16.h>

// ---------------------------------------------------------------------------
// MI455X (gfx1250) implementation:
//   K1: row inverse-norms via V_WMMA_F32_16X16X4_F32 (B = ones -> row sums)
//   K2: JAX threefry2x32(key=42) + Giles erfinv  -> out = noise
//   K3: edge scatter, 1 wave32 per edge; edge triples staged via
//       GLOBAL_LOAD_ASYNC_TO_LDS into interleaved 16B records, invn folded at
//       stage time, consume loop = ds_load_b128 + global_load_b64 + 2 atomics.
// ---------------------------------------------------------------------------

typedef __attribute__((ext_vector_type(2))) float v2f;
typedef __attribute__((ext_vector_type(8))) float v8f;

#ifndef __has_builtin
#define __has_builtin(x) 0
#endif
#if __has_builtin(__builtin_amdgcn_global_load_async_to_lds_b32)
#define NAP_ASYNC_LDS 1
#else
#define NAP_ASYNC_LDS 0
#endif

typedef __attribute__((address_space(1))) int nap_gi32;   // global int
typedef __attribute__((address_space(3))) int nap_li32;   // LDS int

#define NAP_D 64

// ---------------- Kernel 1: inverse L2 norms via WMMA ----------------------
__global__ void __launch_bounds__(256) nap_invnorm_wmma(
    const float* __restrict__ x, float* __restrict__ invn, int N)
{
    const int lane = threadIdx.x & 31;
    const int wv   = threadIdx.x >> 5;
    const int rowBase = (blockIdx.x * 8 + wv) * 16;   // 16 rows per wave
    if (rowBase >= N) return;

    if (rowBase + 16 <= N) {
        // A-matrix 16x4 f32 layout: lanes 0-15 -> M=0..15 with K={k0,k0+1},
        // lanes 16-31 -> same M with K={k0+2,k0+3}.
        const int m  = lane & 15;
        const int kh = (lane >> 4) * 2;
        const float* rp = x + (size_t)(rowBase + m) * NAP_D + kh;

        v8f c = {};                 // accumulator (16x16 f32 C/D tile)
        v2f ones; ones.x = 1.0f; ones.y = 1.0f;   // B = all ones (layout-invariant)
#pragma unroll
        for (int kw = 0; kw < 16; ++kw) {         // 16 * K4 = K64
            v2f v = *(const v2f*)(rp + kw * 4);
            v2f a = v * v;
            c = __builtin_amdgcn_wmma_f32_16x16x4_f32(
                    false, a, false, ones, (short)0, c, false, false);
        }
        // C/D layout: VGPR r, lanes 0-15 hold M=r, lanes 16-31 hold M=r+8;
        // every N-column is identical (B==1), so read column 0 / 16.
        if (m == 0) {
            const int rb = rowBase + (lane >> 4) * 8;
#pragma unroll
            for (int r = 0; r < 8; ++r)
                invn[rb + r] = 1.0f / fmaxf(sqrtf(c[r]), 1e-12f);
        }
    } else {
        // generic tail (not hit for N=100000): one row per lane
        const int r = rowBase + lane;
        if (r < N) {
            const float* rp = x + (size_t)r * NAP_D;
            float s = 0.0f;
            for (int k = 0; k < NAP_D; ++k) s += rp[k] * rp[k];
            invn[r] = 1.0f / fmaxf(sqrtf(s), 1e-12f);
        }
    }
}

// ---------------- Kernel 2: JAX-compatible Gaussian noise ------------------
__device__ __forceinline__ unsigned nap_rotl(unsigned v, int n) {
    return (v << n) | (v >> (32 - n));
}

__device__ __forceinline__ void nap_threefry2x32(
    unsigned c0, unsigned c1, unsigned& o0, unsigned& o1)
{
    const unsigned ks0 = 0u;                      // key(42) -> (hi=0, lo=42)
    const unsigned ks1 = 42u;
    const unsigned ks2 = 0x1BD11BDAu ^ ks0 ^ ks1;
    unsigned x0 = c0 + ks0, x1 = c1 + ks1;
#define NAP_TFR(r) { x0 += x1; x1 = nap_rotl(x1, r); x1 ^= x0; }
    NAP_TFR(13) NAP_TFR(15) NAP_TFR(26) NAP_TFR(6)   x0 += ks1; x1 += ks2 + 1u;
    NAP_TFR(17) NAP_TFR(29) NAP_TFR(16) NAP_TFR(24)  x0 += ks2; x1 += ks0 + 2u;
    NAP_TFR(13) NAP_TFR(15) NAP_TFR(26) NAP_TFR(6)   x0 += ks0; x1 += ks1 + 3u;
    NAP_TFR(17) NAP_TFR(29) NAP_TFR(16) NAP_TFR(24)  x0 += ks1; x1 += ks2 + 4u;
    NAP_TFR(13) NAP_TFR(15) NAP_TFR(26) NAP_TFR(6)   x0 += ks2; x1 += ks0 + 5u;
#undef NAP_TFR
    o0 = x0; o1 = x1;
}

__device__ __forceinline__ float nap_bits_to_normal(unsigned bits) {
    // JAX uniform(lo=nextafter(-1,0), hi=1): mantissa trick then affine map
    const float lo = -0.99999994f;
    const float span = 1.99999994f;
    unsigned fb = (bits >> 9) | 0x3f800000u;
    float f = __uint_as_float(fb) - 1.0f;        // [0,1)
    float u = fmaxf(lo, f * span + lo);
    // erfinv, Giles 2012 single precision (matches XLA ErfInv f32)
    float w = -log1pf(-u * u);
    float p;
    if (w < 5.0f) {
        w = w - 2.5f;
        p =              2.81022636e-08f;
        p = fmaf(p, w,   3.43273939e-07f);
        p = fmaf(p, w,  -3.5233877e-06f);
        p = fmaf(p, w,  -4.39150654e-06f);
        p = fmaf(p, w,   0.00021858087f);
        p = fmaf(p, w,  -0.00125372503f);
        p = fmaf(p, w,  -0.00417768164f);
        p = fmaf(p, w,   0.246640727f);
        p = fmaf(p, w,   1.50140941f);
    } else {
        w = sqrtf(w) - 3.0f;
        p =             -0.000200214257f;
        p = fmaf(p, w,   0.000100950558f);
        p = fmaf(p, w,   0.00134934322f);
        p = fmaf(p, w,  -0.00367342844f);
        p = fmaf(p, w,   0.00573950773f);
        p = fmaf(p, w,  -0.0076224613f);
        p = fmaf(p, w,   0.00943887047f);
        p = fmaf(p, w,   1.00167406f);
        p = fmaf(p, w,   2.83297682f);
    }
    return 1.41421356237f * (p * u);             // sqrt(2)*erfinv(u); NOISE_STD=1
}

__global__ void __launch_bounds__(256) nap_noise_init(
    float* __restrict__ out, int half)
{
    const int i = blockIdx.x * 256 + threadIdx.x;
    if (i >= half) return;
    unsigned b0, b1;
    nap_threefry2x32((unsigned)i, (unsigned)(i + half), b0, b1);
    out[i]        = nap_bits_to_normal(b0);
    out[i + half] = nap_bits_to_normal(b1);
}

// ---------------- Kernel 3: edge scatter (SpMM) ----------------------------
__global__ void __launch_bounds__(256) nap_scatter(
    const float* __restrict__ x, const int* __restrict__ row,
    const int* __restrict__ col, const float* __restrict__ vals,
    const float* __restrict__ invn, float* __restrict__ out, int E)
{
    // Interleaved 16B records: {row, col, val (later folded), pad}
    __shared__ int s_rec[256][4];

    const int tid   = threadIdx.x;
    const int tbase = blockIdx.x * 256;
    const int g     = tbase + tid;

    if (g < E) {
#if NAP_ASYNC_LDS
        __builtin_amdgcn_global_load_async_to_lds_b32(
            (nap_gi32*)(row + g), (nap_li32*)&s_rec[tid][0], 0, 0);
        __builtin_amdgcn_global_load_async_to_lds_b32(
            (nap_gi32*)(col + g), (nap_li32*)&s_rec[tid][1], 0, 0);
        __builtin_amdgcn_global_load_async_to_lds_b32(
            (nap_gi32*)(vals + g), (nap_li32*)&s_rec[tid][2], 0, 0);
#else
        s_rec[tid][0] = row[g];
        s_rec[tid][1] = col[g];
        s_rec[tid][2] = __float_as_int(vals[g]);
#endif
    }
#if NAP_ASYNC_LDS
    asm volatile("s_wait_asynccnt 0" ::: "memory");
#endif
    __syncthreads();

    // Fold vals[e] * 1/||x_col[e]|| once per edge (256 concurrent gathers).
    if (g < E) {
        const int   c = s_rec[tid][1];
        const float v = __int_as_float(s_rec[tid][2]);
        s_rec[tid][2] = __float_as_int(v * invn[c]);
    }
    __syncthreads();

    const int nE   = min(256, E - tbase);
    const int wave = tid >> 5;
    const int lane = tid & 31;
    const int beg  = wave * 32;
    const int end  = min(beg + 32, nE);

    // One wave per edge: lane l covers features {2l, 2l+1} (coalesced 256B).
    for (int i = beg; i < end; ++i) {
        const int4 rec = *(const int4*)&s_rec[i][0];   // ds_load_b128 broadcast
        const int   r = rec.x;
        const int   c = rec.y;
        const float s = __int_as_float(rec.z);
        const v2f  xv = *(const v2f*)(x + (size_t)c * NAP_D + lane * 2);
        float* dst = out + (size_t)r * NAP_D + lane * 2;
        __hip_atomic_fetch_add(dst,     xv.x * s, __ATOMIC_RELAXED,
                               __HIP_MEMORY_SCOPE_AGENT);
        __hip_atomic_fetch_add(dst + 1, xv.y * s, __ATOMIC_RELAXED,
                               __HIP_MEMORY_SCOPE_AGENT);
    }
}

// ---------------------------------------------------------------------------
extern "C" void kernel_launch(void* const* d_in, const int* in_sizes, int n_in,
                              void* d_out, int out_size, void* d_ws, size_t ws_size,
                              hipStream_t stream) {
    const float* x    = (const float*)d_in[0];
    const int*   row  = (const int*)d_in[1];
    const int*   col  = (const int*)d_in[2];
    const float* vals = (const float*)d_in[3];
    float*       out  = (float*)d_out;
    float*       invn = (float*)d_ws;             // N floats of scratch

    const int N = in_sizes[0] / NAP_D;            // 100000
    const int E = in_sizes[1];                    // 1600000

    // K1: inverse norms (8 waves/block * 16 rows = 128 rows per block)
    const int blocksA = (N + 127) / 128;
    nap_invnorm_wmma<<<blocksA, 256, 0, stream>>>(x, invn, N);

    // K2: initialize out with exact JAX gaussian noise
    const int half = (N * NAP_D) / 2;
    nap_noise_init<<<(half + 255) / 256, 256, 0, stream>>>(out, half);

    // K3: scatter-add messages
    nap_scatter<<<(E + 255) / 256, 256, 0, stream>>>(x, row, col, vals, invn, out, E);
}